// SmoothL1Loss_67637144978192
// MI455X (gfx1250) — compile-verified
//
#include <hip/hip_runtime.h>
#include <cstdint>

typedef float v2f __attribute__((ext_vector_type(2)));
typedef float v8f __attribute__((ext_vector_type(8)));

#define TJ  128   // targets per block tile
#define BLK 256   // threads per block (8 wave32)

// Axis-aligned hull of a rotated box, matching rotbox2polys+poly2bbox:
// the 4 corners cover all sign combos of (cs*hw, ss*hh), so
// min/max reduce to center +/- (|cs|*hw + |ss|*hh) etc.
__device__ __forceinline__ void hbb_from_box(float b0, float b1, float b2, float b3, float b4,
                                             float& x1, float& y1, float& x2, float& y2,
                                             float& area)
{
    float w  = b2 - 1.0f;
    float h  = b3 - 1.0f;
    float cs = cosf(b4);
    float ss = sinf(b4);
    float hw = 0.5f * w;
    float hh = 0.5f * h;
    float ex = fabsf(cs) * hw + fabsf(ss) * hh;
    float ey = fabsf(ss) * hw + fabsf(cs) * hh;
    x1 = b0 - ex; x2 = b0 + ex;
    y1 = b1 - ey; y2 = b1 + ey;
    area = (x2 - x1 + 1.0f) * (y2 - y1 + 1.0f);
}

// Sum x over all 32 lanes of a wave, returned to every lane.
// Uses V_WMMA_F32_16X16X4_F32 with A = per-lane partials, B = ones:
//   A layout (ISA 7.12.2): lane L<16  -> A[L][0]=v0, A[L][1]=v1
//                          lane L>=16 -> A[L-16][2]=v0, A[L-16][3]=v1
//   => D[m][n] = x_m + x_{m+16}  (B, C constant => layout-independent)
//   D layout: lane n<16 vgpr p = D[p][n]; lane n>=16 vgpr p = D[p+8][n]
//   => per-lane sum of 8 D regs + shfl_xor(16) = full 32-lane sum.
__device__ __forceinline__ float wave_sum32(float x)
{
#if __has_builtin(__builtin_amdgcn_wmma_f32_16x16x4_f32)
    v2f a; a[0] = x;    a[1] = 0.0f;
    v2f b; b[0] = 1.0f; b[1] = 1.0f;
    v8f c = {};
    v8f d = __builtin_amdgcn_wmma_f32_16x16x4_f32(false, a, false, b,
                                                  (short)0, c, false, false);
    float s = ((d[0] + d[1]) + (d[2] + d[3])) + ((d[4] + d[5]) + (d[6] + d[7]));
    s += __shfl_xor(s, 16, 32);
    return s;
#else
    for (int off = 16; off > 0; off >>= 1) x += __shfl_xor(x, off, 32);
    return x;
#endif
}

__global__ void zero_kernel(float* __restrict__ o, int n)
{
    int i = blockIdx.x * blockDim.x + threadIdx.x;
    if (i < n) o[i] = 0.0f;
}

__global__ __launch_bounds__(BLK)
void pair_kernel(const float* __restrict__ pred,
                 const float* __restrict__ tgt,
                 float* __restrict__ out,
                 int np, int nt)
{
    __shared__ __align__(16) float  raw[TJ * 5];  // raw target tile (async-loaded)
    __shared__ __align__(16) float2 rec[TJ * 5];  // records: (x1,y1)(x2,y2)(area,b0)(b1,b2)(b3,b4)
    __shared__ float bsum[5];

    const int tid = threadIdx.x;
    const int i   = blockIdx.x * BLK + tid;
    const int j0  = blockIdx.y * TJ;
    const int jc  = min(TJ, nt - j0);

    // ---- stage raw target tile into LDS via CDNA5 async copy ----
    {
        const int nbytes = jc * 5 * 4;
        const int nchunk = (nbytes + 15) >> 4;           // 16B chunks
        const uint32_t lds0 = (uint32_t)(uintptr_t)(&raw[0]);
        const unsigned long long g0 =
            (unsigned long long)(uintptr_t)(tgt + (size_t)j0 * 5);
        for (int c = tid; c < nchunk; c += BLK) {
            uint32_t           loff = lds0 + (uint32_t)(c << 4);
            unsigned long long ga   = g0 + (unsigned long long)(c << 4);
            asm volatile("global_load_async_to_lds_b128 %0, %1, off"
                         :: "v"(loff), "v"(ga) : "memory");
        }
#if __has_builtin(__builtin_amdgcn_s_wait_asynccnt)
        __builtin_amdgcn_s_wait_asynccnt(0);
#else
        asm volatile("s_wait_asynccnt 0" ::: "memory");
#endif
    }
    if (tid < 5) bsum[tid] = 0.0f;
    __syncthreads();

    // ---- expand raw tile into hull records (once per block) ----
    if (tid < jc) {
        float b0 = raw[tid * 5 + 0];
        float b1 = raw[tid * 5 + 1];
        float b2 = raw[tid * 5 + 2];
        float b3 = raw[tid * 5 + 3];
        float b4 = raw[tid * 5 + 4];
        float x1, y1, x2, y2, ar;
        hbb_from_box(b0, b1, b2, b3, b4, x1, y1, x2, y2, ar);
        rec[tid * 5 + 0] = make_float2(x1, y1);
        rec[tid * 5 + 1] = make_float2(x2, y2);
        rec[tid * 5 + 2] = make_float2(ar, b0);
        rec[tid * 5 + 3] = make_float2(b1, b2);
        rec[tid * 5 + 4] = make_float2(b3, b4);
    }
    __syncthreads();

    // ---- per-thread pred record (registers) ----
    const bool valid = (i < np);
    const int  ii    = valid ? i : 0;
    const float p0 = pred[ii * 5 + 0];
    const float p1 = pred[ii * 5 + 1];
    const float p2 = pred[ii * 5 + 2];
    const float p3 = pred[ii * 5 + 3];
    const float p4 = pred[ii * 5 + 4];
    float px1, py1, px2, py2, pa;
    hbb_from_box(p0, p1, p2, p3, p4, px1, py1, px2, py2, pa);
    const float vm = valid ? 1.0f : 0.0f;

    float a0 = 0.0f, a1 = 0.0f, a2 = 0.0f, a3 = 0.0f, a4 = 0.0f;

    for (int j = 0; j < jc; ++j) {
        const float2 r0 = rec[j * 5 + 0];   // x1, y1
        const float2 r1 = rec[j * 5 + 1];   // x2, y2
        const float2 r2 = rec[j * 5 + 2];   // area, b0
        const float2 r3 = rec[j * 5 + 3];   // b1, b2
        const float2 r4 = rec[j * 5 + 4];   // b3, b4
        // IoU of axis-aligned hulls (broadcast LDS reads)
        float ltx = fmaxf(px1, r0.x);
        float lty = fmaxf(py1, r0.y);
        float rbx = fminf(px2, r1.x);
        float rby = fminf(py2, r1.y);
        float wx  = fmaxf(rbx - ltx + 1.0f, 0.0f);
        float wy  = fmaxf(rby - lty + 1.0f, 0.0f);
        float ov  = wx * wy;
        float iou = ov / (pa + r2.x - ov);
        float wgt = (ov > 0.0f) ? -logf(iou) : 0.0f;   // matches where(mask,-log(iou),0)
        wgt *= vm;
        // smooth L1 on raw 5 params (BETA = 1)
        float d0 = fabsf(r2.y - p0);
        float d1 = fabsf(r3.x - p1);
        float d2 = fabsf(r3.y - p2);
        float d3 = fabsf(r4.x - p3);
        float d4 = fabsf(r4.y - p4);
        float l0 = (d0 < 1.0f) ? 0.5f * d0 * d0 : d0 - 0.5f;
        float l1 = (d1 < 1.0f) ? 0.5f * d1 * d1 : d1 - 0.5f;
        float l2 = (d2 < 1.0f) ? 0.5f * d2 * d2 : d2 - 0.5f;
        float l3 = (d3 < 1.0f) ? 0.5f * d3 * d3 : d3 - 0.5f;
        float l4 = (d4 < 1.0f) ? 0.5f * d4 * d4 : d4 - 0.5f;
        float ss2 = ((l0 * l0 + l1 * l1) + (l2 * l2 + l3 * l3)) + l4 * l4;
        float scale = wgt / sqrtf(ss2);
        a0 = fmaf(l0, scale, a0);
        a1 = fmaf(l1, scale, a1);
        a2 = fmaf(l2, scale, a2);
        a3 = fmaf(l3, scale, a3);
        a4 = fmaf(l4, scale, a4);
    }

    // ---- wave reduction via WMMA, then LDS + global accumulation ----
    float t0 = wave_sum32(a0);
    float t1 = wave_sum32(a1);
    float t2 = wave_sum32(a2);
    float t3 = wave_sum32(a3);
    float t4 = wave_sum32(a4);
    if ((tid & 31) == 0) {
        atomicAdd(&bsum[0], t0);
        atomicAdd(&bsum[1], t1);
        atomicAdd(&bsum[2], t2);
        atomicAdd(&bsum[3], t3);
        atomicAdd(&bsum[4], t4);
    }
    __syncthreads();
    if (tid < 5) atomicAdd(&out[tid], bsum[tid]);
}

extern "C" void kernel_launch(void* const* d_in, const int* in_sizes, int n_in,
                              void* d_out, int out_size, void* d_ws, size_t ws_size,
                              hipStream_t stream)
{
    (void)n_in; (void)d_ws; (void)ws_size;
    const float* pred = (const float*)d_in[0];
    const float* tgt  = (const float*)d_in[1];
    float*       out  = (float*)d_out;
    const int np = in_sizes[0] / 5;   // 4096
    const int nt = in_sizes[1] / 5;   // 2048

    zero_kernel<<<1, 64, 0, stream>>>(out, out_size);

    dim3 grid((np + BLK - 1) / BLK, (nt + TJ - 1) / TJ);
    pair_kernel<<<grid, BLK, 0, stream>>>(pred, tgt, out, np, nt);
}